// GAT_Measures_Net_ode_func_53661321396516
// MI455X (gfx1250) — compile-verified
//
#include <hip/hip_runtime.h>
#include <hip/hip_bf16.h>

// GAT 2-layer dense, N=4096, F=256, D=64, H=4.
// Fused flash-style masked softmax; both GEMMs on v_wmma_f32_16x16x4_f32.
// PV kernel: 4-way split-j per workgroup + LDS reduction for occupancy.

#define GN 4096
#define GK 256   // F == H*D == 256 for both layers
#define GD 64
#define GH 4
#define ALPHA_LRELU 0.2f
#define NEGV (-9e15f)

typedef float v2f __attribute__((ext_vector_type(2)));
typedef float v8f __attribute__((ext_vector_type(8)));

// ---------------------------------------------------------------------------
// Kernel 1: per-head feature GEMM  Hout[h][n][d] = sum_k X[n][k] * W[h][k][d]
// One wave per (head, 16-row block). 4 N-tiles of 16, K=256 in steps of 4.
// ---------------------------------------------------------------------------
__global__ __launch_bounds__(32) void feat_gemm(const float* __restrict__ X,   // [N, 256]
                                                const float* __restrict__ W,   // [H, 256, 64]
                                                float* __restrict__ Hout)      // [H, N, 64]
{
    const int head = blockIdx.y;
    const int row0 = blockIdx.x * 16;
    const int lane = threadIdx.x;
    const int li = lane & 15;
    const int lh = lane >> 4;   // selects K pair {0,1} vs {2,3}

    const float* __restrict__ Wh = W + (size_t)head * GK * GD;
    const float* __restrict__ xrow = X + (size_t)(row0 + li) * GK + 2 * lh;

    v8f acc0 = {}, acc1 = {}, acc2 = {}, acc3 = {};

    for (int k = 0; k < GK; k += 4) {
        const v2f a = *(const v2f*)(xrow + k);   // A: X[row0+li][k + 2*lh + {0,1}]
        const float* __restrict__ wp = Wh + (size_t)(k + 2 * lh) * GD + li;
        v2f b0, b1, b2, b3;
        b0.x = wp[0];      b0.y = wp[GD];
        b1.x = wp[16];     b1.y = wp[16 + GD];
        b2.x = wp[32];     b2.y = wp[32 + GD];
        b3.x = wp[48];     b3.y = wp[48 + GD];
        acc0 = __builtin_amdgcn_wmma_f32_16x16x4_f32(false, a, false, b0, (short)0, acc0, false, false);
        acc1 = __builtin_amdgcn_wmma_f32_16x16x4_f32(false, a, false, b1, (short)0, acc1, false, false);
        acc2 = __builtin_amdgcn_wmma_f32_16x16x4_f32(false, a, false, b2, (short)0, acc2, false, false);
        acc3 = __builtin_amdgcn_wmma_f32_16x16x4_f32(false, a, false, b3, (short)0, acc3, false, false);
    }

    float* __restrict__ hp = Hout + (size_t)head * GN * GD + (size_t)row0 * GD;
    v8f accs[4] = {acc0, acc1, acc2, acc3};
#pragma unroll
    for (int t = 0; t < 4; ++t)
#pragma unroll
        for (int r = 0; r < 8; ++r)
            hp[(size_t)(r + 8 * lh) * GD + t * 16 + li] = accs[t][r];
}

// ---------------------------------------------------------------------------
// Kernel 2: attention projections f1[h][n] = h . a[:D], f2[h][n] = h . a[D:]
// ---------------------------------------------------------------------------
__global__ void attn_proj(const float* __restrict__ Hmat,  // [H, N, 64]
                          const float* __restrict__ avec,  // [H, 128]
                          float* __restrict__ f1, float* __restrict__ f2)
{
    const int idx = blockIdx.x * blockDim.x + threadIdx.x;  // over H*N
    if (idx >= GH * GN) return;
    const int head = idx / GN;
    const float* __restrict__ hp = Hmat + (size_t)idx * GD;
    const float* __restrict__ a1 = avec + (size_t)head * 2 * GD;
    const float* __restrict__ a2 = a1 + GD;
    float s1 = 0.f, s2 = 0.f;
#pragma unroll 8
    for (int d = 0; d < GD; ++d) {
        const float v = hp[d];
        s1 += v * a1[d];
        s2 += v * a2[d];
    }
    f1[idx] = s1;
    f2[idx] = s2;
}

// ---------------------------------------------------------------------------
// Kernel 3: per-row softmax stats, two-phase (max pass, then denom pass; the
// second pass hits L2). All 4 heads per row so adj is streamed once from HBM.
// Vectorized int4/float4 loads. One wave per row.
// ---------------------------------------------------------------------------
__global__ __launch_bounds__(32) void row_stats(const int* __restrict__ adj,
                                                const float* __restrict__ f1,
                                                const float* __restrict__ f2,
                                                float* __restrict__ mrow,
                                                float* __restrict__ lrow)
{
    const int i = blockIdx.x;
    const int lane = threadIdx.x;
    float f1i[GH];
#pragma unroll
    for (int h = 0; h < GH; ++h) f1i[h] = f1[h * GN + i];

    const int4* __restrict__ arow4 = (const int4*)(adj + (size_t)i * GN);

    // phase 1: row max of masked leakyrelu(f1_i + f2_j)
    float m[GH];
#pragma unroll
    for (int h = 0; h < GH; ++h) m[h] = NEGV;
    for (int j4 = lane; j4 < GN / 4; j4 += 32) {
        const int4 a4 = arow4[j4];
#pragma unroll
        for (int h = 0; h < GH; ++h) {
            const float4 g = *(const float4*)(f2 + h * GN + 4 * j4);
            float e0 = f1i[h] + g.x, e1 = f1i[h] + g.y, e2 = f1i[h] + g.z, e3 = f1i[h] + g.w;
            e0 = (e0 > 0.f) ? e0 : ALPHA_LRELU * e0;
            e1 = (e1 > 0.f) ? e1 : ALPHA_LRELU * e1;
            e2 = (e2 > 0.f) ? e2 : ALPHA_LRELU * e2;
            e3 = (e3 > 0.f) ? e3 : ALPHA_LRELU * e3;
            e0 = (a4.x > 0) ? e0 : NEGV;
            e1 = (a4.y > 0) ? e1 : NEGV;
            e2 = (a4.z > 0) ? e2 : NEGV;
            e3 = (a4.w > 0) ? e3 : NEGV;
            m[h] = fmaxf(m[h], fmaxf(fmaxf(e0, e1), fmaxf(e2, e3)));
        }
    }
#pragma unroll
    for (int off = 16; off > 0; off >>= 1)
#pragma unroll
        for (int h = 0; h < GH; ++h) m[h] = fmaxf(m[h], __shfl_xor(m[h], off, 32));

    // phase 2: denominator sum(exp(e - m))
    float l[GH];
#pragma unroll
    for (int h = 0; h < GH; ++h) l[h] = 0.f;
    for (int j4 = lane; j4 < GN / 4; j4 += 32) {
        const int4 a4 = arow4[j4];
#pragma unroll
        for (int h = 0; h < GH; ++h) {
            const float4 g = *(const float4*)(f2 + h * GN + 4 * j4);
            float e0 = f1i[h] + g.x, e1 = f1i[h] + g.y, e2 = f1i[h] + g.z, e3 = f1i[h] + g.w;
            e0 = (e0 > 0.f) ? e0 : ALPHA_LRELU * e0;
            e1 = (e1 > 0.f) ? e1 : ALPHA_LRELU * e1;
            e2 = (e2 > 0.f) ? e2 : ALPHA_LRELU * e2;
            e3 = (e3 > 0.f) ? e3 : ALPHA_LRELU * e3;
            e0 = (a4.x > 0) ? e0 : NEGV;
            e1 = (a4.y > 0) ? e1 : NEGV;
            e2 = (a4.z > 0) ? e2 : NEGV;
            e3 = (a4.w > 0) ? e3 : NEGV;
            l[h] += __expf(e0 - m[h]) + __expf(e1 - m[h]) + __expf(e2 - m[h]) + __expf(e3 - m[h]);
        }
    }
#pragma unroll
    for (int off = 16; off > 0; off >>= 1)
#pragma unroll
        for (int h = 0; h < GH; ++h) l[h] += __shfl_xor(l[h], off, 32);

    if (lane == 0) {
#pragma unroll
        for (int h = 0; h < GH; ++h) {
            mrow[h * GN + i] = m[h];
            lrow[h * GN + i] = l[h];
        }
    }
}

// ---------------------------------------------------------------------------
// Kernel 4: fused attention * h with WMMA. 128 threads = 4 waves per
// (head, 16-row block); each wave accumulates P @ h over a disjoint quarter
// of the j range (partials sum exactly since m_i is global), then an LDS
// reduction + shared ELU epilogue combines them.
// ---------------------------------------------------------------------------
__global__ __launch_bounds__(128) void attn_pv(const int* __restrict__ adj,
                                               const float* __restrict__ Hmat,  // [H, N, 64]
                                               const float* __restrict__ f1,
                                               const float* __restrict__ f2,
                                               const float* __restrict__ mrow,
                                               const float* __restrict__ lrow,
                                               float* __restrict__ out)         // [N, 256]
{
    __shared__ float red[4 * 16 * GD];   // 16 KB: one 16x64 partial per wave

    const int head = blockIdx.y;
    const int row0 = blockIdx.x * 16;
    const int tid = threadIdx.x;
    const int wave = tid >> 5;           // 0..3 -> j quarter
    const int lane = tid & 31;
    const int li = lane & 15;
    const int lh = (lane >> 4) & 1;

    // per-lane row constants (A-fragment row is li)
    const float f1m = f1[head * GN + row0 + li];
    const float mi = mrow[head * GN + row0 + li];

    const int* __restrict__ arow = adj + (size_t)(row0 + li) * GN + 2 * lh;
    const float* __restrict__ f2h = f2 + head * GN + 2 * lh;
    const float* __restrict__ Hh = Hmat + (size_t)head * GN * GD;

    v8f acc0 = {}, acc1 = {}, acc2 = {}, acc3 = {};

    const int jbeg = wave * (GN / 4);
    const int jend = jbeg + (GN / 4);

    for (int j0 = jbeg; j0 < jend; j0 += 16) {
        __builtin_prefetch(arow + j0 + 128, 0, 0);   // stream adj row ahead
        // this lane's 8 P entries of the 16x16 block: cols j0 + 2*lh + {0,1} + 4q
        float p[8];
#pragma unroll
        for (int q = 0; q < 4; ++q) {
            const int c = j0 + 4 * q;
            const int a0 = arow[c];
            const int a1v = arow[c + 1];
            float e0 = f1m + f2h[c];
            float e1 = f1m + f2h[c + 1];
            e0 = (e0 > 0.f) ? e0 : ALPHA_LRELU * e0;
            e1 = (e1 > 0.f) ? e1 : ALPHA_LRELU * e1;
            e0 = (a0 > 0) ? e0 : NEGV;
            e1 = (a1v > 0) ? e1 : NEGV;
            p[2 * q] = __expf(e0 - mi);
            p[2 * q + 1] = __expf(e1 - mi);
        }
#pragma unroll
        for (int q = 0; q < 4; ++q) {
            v2f a;
            a.x = p[2 * q];
            a.y = p[2 * q + 1];
            const float* __restrict__ hb = Hh + (size_t)(j0 + 4 * q + 2 * lh) * GD + li;
            v2f b0, b1, b2, b3;
            b0.x = hb[0];   b0.y = hb[GD];
            b1.x = hb[16];  b1.y = hb[16 + GD];
            b2.x = hb[32];  b2.y = hb[32 + GD];
            b3.x = hb[48];  b3.y = hb[48 + GD];
            acc0 = __builtin_amdgcn_wmma_f32_16x16x4_f32(false, a, false, b0, (short)0, acc0, false, false);
            acc1 = __builtin_amdgcn_wmma_f32_16x16x4_f32(false, a, false, b1, (short)0, acc1, false, false);
            acc2 = __builtin_amdgcn_wmma_f32_16x16x4_f32(false, a, false, b2, (short)0, acc2, false, false);
            acc3 = __builtin_amdgcn_wmma_f32_16x16x4_f32(false, a, false, b3, (short)0, acc3, false, false);
        }
    }

    // spill this wave's 16x64 partial into LDS (C layout: row = r+8*lh, col = t*16+li)
    float* __restrict__ rp = red + wave * (16 * GD);
    v8f accs[4] = {acc0, acc1, acc2, acc3};
#pragma unroll
    for (int t = 0; t < 4; ++t)
#pragma unroll
        for (int r = 0; r < 8; ++r)
            rp[(r + 8 * lh) * GD + t * 16 + li] = accs[t][r];
    __syncthreads();

    // cooperative reduce + softmax normalize + ELU + store (1024 elems / 128 thr)
#pragma unroll
    for (int e = 0; e < 8; ++e) {
        const int idx = tid + e * 128;
        const int row = idx >> 6;       // 0..15
        const int col = idx & 63;
        float v = red[idx] + red[idx + 1024] + red[idx + 2048] + red[idx + 3072];
        v *= 1.0f / lrow[head * GN + row0 + row];
        v = (v > 0.f) ? v : (__expf(v) - 1.0f);
        out[(size_t)(row0 + row) * (GH * GD) + head * GD + col] = v;
    }
}

// ---------------------------------------------------------------------------
extern "C" void kernel_launch(void* const* d_in, const int* in_sizes, int n_in,
                              void* d_out, int out_size, void* d_ws, size_t ws_size,
                              hipStream_t stream)
{
    // setup_inputs order: t, x, adj, W1, a1, W2, a2
    const float* x   = (const float*)d_in[1];   // [4096, 256]
    const int*   adj = (const int*)d_in[2];     // [4096, 4096]
    const float* W1  = (const float*)d_in[3];   // [4, 256, 64]
    const float* a1  = (const float*)d_in[4];   // [4, 128]
    const float* W2  = (const float*)d_in[5];   // [4, 256, 64]
    const float* a2  = (const float*)d_in[6];   // [4, 128]
    float* out = (float*)d_out;                 // [4096, 256]

    float* h_buf = (float*)d_ws;                // H*N*D = 1,048,576 floats
    float* f1    = h_buf + (size_t)GH * GN * GD;
    float* f2    = f1 + GH * GN;
    float* mrow  = f2 + GH * GN;
    float* lrow  = mrow + GH * GN;
    float* h1    = lrow + GH * GN;              // layer-1 output [N, 256]

    const dim3 gGemm(GN / 16, GH);

    // ---- layer 1 ----
    feat_gemm<<<gGemm, 32, 0, stream>>>(x, W1, h_buf);
    attn_proj<<<(GH * GN + 255) / 256, 256, 0, stream>>>(h_buf, a1, f1, f2);
    row_stats<<<GN, 32, 0, stream>>>(adj, f1, f2, mrow, lrow);
    attn_pv<<<gGemm, 128, 0, stream>>>(adj, h_buf, f1, f2, mrow, lrow, h1);

    // ---- layer 2 ----
    feat_gemm<<<gGemm, 32, 0, stream>>>(h1, W2, h_buf);
    attn_proj<<<(GH * GN + 255) / 256, 256, 0, stream>>>(h_buf, a2, f1, f2);
    row_stats<<<GN, 32, 0, stream>>>(adj, f1, f2, mrow, lrow);
    attn_pv<<<gGemm, 128, 0, stream>>>(adj, h_buf, f1, f2, mrow, lrow, out);
}